// TGCN_76381698392139
// MI455X (gfx1250) — compile-verified
//
#include <hip/hip_runtime.h>
#include <hip/hip_bf16.h>

// ---------------------------------------------------------------------------
// GConvGRU (ChebConv K=2 GRU) for MI455X / gfx1250.
//   B=1, T=128, N=50000, C_IN=16, C_HID=32, C_OUT=8, E=1.6M
// Dense gate GEMMs use V_WMMA_F32_16X16X4_F32 (fp32 WMMA, wave32).
// Edge aggregation = thread-per-(edge, 4ch) with global_atomic_add_f32;
// all hot buffers (h, x_t, idx, w) are L2-resident (192MB L2).
// ---------------------------------------------------------------------------

typedef __attribute__((ext_vector_type(2))) float v2f;
typedef __attribute__((ext_vector_type(8))) float v8f;

static constexpr int TT   = 128;
static constexpr int NN   = 50000;
static constexpr int CIN  = 16;
static constexpr int CH   = 32;
static constexpr int COUT = 8;
static constexpr int NTILE = NN / 16;   // 3125 node tiles of 16

__device__ __forceinline__ v8f wmma4(v2f a, v2f b, v8f c) {
  // D = A(16x4 f32) * B(4x16 f32) + C(16x16 f32)
  return __builtin_amdgcn_wmma_f32_16x16x4_f32(
      /*neg_a=*/false, a, /*neg_b=*/false, b,
      /*c_mod=*/(short)0, c, /*reuse_a=*/false, /*reuse_b=*/false);
}

// Accumulate acc += A[16 x 4*K4] @ W[., ldw] (cols [colbase, colbase+16),
// zero-padded past ncols). Arow0 points at row 0 of the 16-row tile.
template <int K4>
__device__ __forceinline__ void mm_acc(const float* __restrict__ Arow0, int lda,
                                       const float* __restrict__ W, int ldw,
                                       int ncols, int colbase, int lane,
                                       v8f& acc) {
  const int row   = lane & 15;
  const int khalf = (lane >> 4) << 1;        // lanes 16..31 hold K+2,K+3
  const int col   = colbase + (lane & 15);
  const bool cv   = (col < ncols);
#pragma unroll
  for (int kc = 0; kc < K4; ++kc) {
    const int kb = (kc << 2) + khalf;
    v2f a, b;
    a.x = Arow0[row * lda + kb];
    a.y = Arow0[row * lda + kb + 1];
    b.x = cv ? W[kb * ldw + col] : 0.0f;
    b.y = cv ? W[(kb + 1) * ldw + col] : 0.0f;
    acc = wmma4(a, b, acc);
  }
}

__device__ __forceinline__ float sigm(float x) {
  return 1.0f / (1.0f + __expf(-x));
}

// ------------------------------ utility kernels ----------------------------

__global__ void k_zero(float* __restrict__ p, int n) {
  int i = blockIdx.x * blockDim.x + threadIdx.x;
  if (i < n) p[i] = 0.0f;
}

__global__ void k_deg(const int* __restrict__ src, float* __restrict__ deg, int E) {
  int e = blockIdx.x * blockDim.x + threadIdx.x;
  if (e < E) atomicAdd(&deg[src[e]], 1.0f);
}

__global__ void k_dinv(float* __restrict__ deg, int n) {
  int i = blockIdx.x * blockDim.x + threadIdx.x;
  if (i < n) {
    float d = deg[i];
    deg[i] = (d > 0.0f) ? rsqrtf(d < 1.0f ? 1.0f : d) : 0.0f;
  }
}

__global__ void k_edgew(const int* __restrict__ src, const int* __restrict__ dst,
                        const float* __restrict__ dinv, float* __restrict__ w, int E) {
  int e = blockIdx.x * blockDim.x + threadIdx.x;
  if (e < E) w[e] = -dinv[src[e]] * dinv[dst[e]];
}

// out[dst] += w[e] * in[src]  over 4 channels per thread (gridDim.y = C/4)
__global__ void k_agg(const int* __restrict__ src, const int* __restrict__ dst,
                      const float* __restrict__ w, const float* __restrict__ in,
                      float* __restrict__ out, int C, int E) {
  int e = blockIdx.x * blockDim.x + threadIdx.x;
  if (e >= E) return;
  const int cg = blockIdx.y << 2;
  const int s = src[e], d = dst[e];
  const float we = w[e];
  const float4 v = *reinterpret_cast<const float4*>(in + (size_t)s * C + cg);
  float* o = out + (size_t)d * C + cg;
  atomicAdd(o + 0, we * v.x);
  atomicAdd(o + 1, we * v.y);
  atomicAdd(o + 2, we * v.z);
  atomicAdd(o + 3, we * v.w);
}

// ------------------------------ gate kernels (WMMA) ------------------------

// Z = sigmoid(cheb(x) + cheb(h)); R likewise; store Z and h*R.
__global__ void k_gate_zr(const float* __restrict__ Xt, const float* __restrict__ Ax,
                          const float* __restrict__ h,  const float* __restrict__ Ah,
                          const float* __restrict__ Wxz, const float* __restrict__ bxz,
                          const float* __restrict__ Whz, const float* __restrict__ bhz,
                          const float* __restrict__ Wxr, const float* __restrict__ bxr,
                          const float* __restrict__ Whr, const float* __restrict__ bhr,
                          float* __restrict__ Zb, float* __restrict__ HRb) {
  const int gtid = blockIdx.x * blockDim.x + threadIdx.x;
  const int tile = gtid >> 5;
  const int lane = threadIdx.x & 31;
  if (tile >= NTILE) return;            // whole-wave uniform: EXEC stays all-1s
  const int base = tile << 4;
  const float* xr  = Xt + (size_t)base * CIN;
  const float* axr = Ax + (size_t)base * CIN;
  const float* hr  = h  + (size_t)base * CH;
  const float* ahr = Ah + (size_t)base * CH;
#pragma unroll
  for (int nt = 0; nt < 2; ++nt) {
    const int cb = nt << 4;
    v8f az = {}; v8f ar = {};
    mm_acc<4>(xr,  CIN, Wxz,            CH, CH, cb, lane, az);
    mm_acc<4>(axr, CIN, Wxz + CIN * CH, CH, CH, cb, lane, az);
    mm_acc<8>(hr,  CH,  Whz,            CH, CH, cb, lane, az);
    mm_acc<8>(ahr, CH,  Whz + CH * CH,  CH, CH, cb, lane, az);
    mm_acc<4>(xr,  CIN, Wxr,            CH, CH, cb, lane, ar);
    mm_acc<4>(axr, CIN, Wxr + CIN * CH, CH, CH, cb, lane, ar);
    mm_acc<8>(hr,  CH,  Whr,            CH, CH, cb, lane, ar);
    mm_acc<8>(ahr, CH,  Whr + CH * CH,  CH, CH, cb, lane, ar);
    const int col = cb + (lane & 15);
    const float bz = bxz[col] + bhz[col];
    const float br = bxr[col] + bhr[col];
    const int mofs = (lane >> 4) << 3;  // C/D: lanes 16..31 hold rows M+8
#pragma unroll
    for (int v = 0; v < 8; ++v) {
      const int node = base + mofs + v;
      const float z  = sigm(az[v] + bz);
      const float r  = sigm(ar[v] + br);
      const float hv = h[(size_t)node * CH + col];
      Zb[(size_t)node * CH + col]  = z;
      HRb[(size_t)node * CH + col] = hv * r;
    }
  }
}

// Ht = tanh(cheb(x) + cheb(h*R)); h = Z*h + (1-Z)*Ht  (in place, disjoint lanes)
__global__ void k_gate_h(const float* __restrict__ Xt, const float* __restrict__ Ax,
                         float* __restrict__ h, const float* __restrict__ HRb,
                         const float* __restrict__ Ahr,
                         const float* __restrict__ Wxh, const float* __restrict__ bxh,
                         const float* __restrict__ Whh, const float* __restrict__ bhh,
                         const float* __restrict__ Zb) {
  const int gtid = blockIdx.x * blockDim.x + threadIdx.x;
  const int tile = gtid >> 5;
  const int lane = threadIdx.x & 31;
  if (tile >= NTILE) return;
  const int base = tile << 4;
  const float* xr   = Xt  + (size_t)base * CIN;
  const float* axr  = Ax  + (size_t)base * CIN;
  const float* hrr  = HRb + (size_t)base * CH;
  const float* ahrr = Ahr + (size_t)base * CH;
#pragma unroll
  for (int nt = 0; nt < 2; ++nt) {
    const int cb = nt << 4;
    v8f ac = {};
    mm_acc<4>(xr,   CIN, Wxh,            CH, CH, cb, lane, ac);
    mm_acc<4>(axr,  CIN, Wxh + CIN * CH, CH, CH, cb, lane, ac);
    mm_acc<8>(hrr,  CH,  Whh,            CH, CH, cb, lane, ac);
    mm_acc<8>(ahrr, CH,  Whh + CH * CH,  CH, CH, cb, lane, ac);
    const int col = cb + (lane & 15);
    const float bb = bxh[col] + bhh[col];
    const int mofs = (lane >> 4) << 3;
#pragma unroll
    for (int v = 0; v < 8; ++v) {
      const int node = base + mofs + v;
      const float ht = tanhf(ac[v] + bb);
      const float z  = Zb[(size_t)node * CH + col];
      const float hv = h[(size_t)node * CH + col];
      h[(size_t)node * CH + col] = z * hv + (1.0f - z) * ht;
    }
  }
}

// out_t = h @ fc_w + fc_b   ([N,32]@[32,8]; B zero-padded to 16 cols)
__global__ void k_fc(const float* __restrict__ h, const float* __restrict__ fcw,
                     const float* __restrict__ fcb, float* __restrict__ out) {
  const int gtid = blockIdx.x * blockDim.x + threadIdx.x;
  const int tile = gtid >> 5;
  const int lane = threadIdx.x & 31;
  if (tile >= NTILE) return;
  const int base = tile << 4;
  v8f ac = {};
  mm_acc<8>(h + (size_t)base * CH, CH, fcw, COUT, COUT, 0, lane, ac);
  const int col = lane & 15;
  if (col < COUT) {                    // divergence only after the WMMAs
    const float bb = fcb[col];
    const int mofs = (lane >> 4) << 3;
#pragma unroll
    for (int v = 0; v < 8; ++v)
      out[(size_t)(base + mofs + v) * COUT + col] = ac[v] + bb;
  }
}

// ------------------------------ host driver --------------------------------

extern "C" void kernel_launch(void* const* d_in, const int* in_sizes, int n_in,
                              void* d_out, int out_size, void* d_ws, size_t ws_size,
                              hipStream_t stream) {
  const float* X   = (const float*)d_in[0];
  const int*   ei  = (const int*)d_in[1];
  const float* Wxz = (const float*)d_in[2];
  const float* bxz = (const float*)d_in[3];
  const float* Whz = (const float*)d_in[4];
  const float* bhz = (const float*)d_in[5];
  const float* Wxr = (const float*)d_in[6];
  const float* bxr = (const float*)d_in[7];
  const float* Whr = (const float*)d_in[8];
  const float* bhr = (const float*)d_in[9];
  const float* Wxh = (const float*)d_in[10];
  const float* bxh = (const float*)d_in[11];
  const float* Whh = (const float*)d_in[12];
  const float* bhh = (const float*)d_in[13];
  const float* fcw = (const float*)d_in[14];
  const float* fcb = (const float*)d_in[15];
  float* out = (float*)d_out;
  (void)n_in; (void)out_size; (void)ws_size;

  const int E = in_sizes[1] / 2;
  const int* src = ei;
  const int* dst = ei + E;

  // ---- workspace carve (256B aligned), ~42 MB total ----
  char* w8 = (char*)d_ws;
  size_t off = 0;
  auto carve = [&](size_t nflt) -> float* {
    float* p = (float*)(w8 + off);
    off += ((nflt * sizeof(float) + 255) / 256) * 256;
    return p;
  };
  float* deg  = carve(NN);                        // later holds dinv
  float* ew   = carve((size_t)E);                 // edge weights
  float* h    = carve((size_t)NN * CH);           // hidden state
  float* AxAh = carve((size_t)NN * (CIN + CH));   // Ax | Ah contiguous
  float* Ax   = AxAh;
  float* Ah   = AxAh + (size_t)NN * CIN;
  float* Zb   = carve((size_t)NN * CH);
  float* HRb  = carve((size_t)NN * CH);
  float* Ahr  = carve((size_t)NN * CH);

  auto zerol = [&](float* p, size_t n) {
    k_zero<<<dim3((unsigned)((n + 255) / 256)), 256, 0, stream>>>(p, (int)n);
  };

  const dim3 eblk((E + 255) / 256);
  const dim3 ggrid((NTILE + 3) / 4);   // 128 threads = 4 wave32 = 4 node tiles
  const int GTPB = 128;

  // ---- graph Laplacian edge weights: w = -dinv[src]*dinv[dst] ----
  zerol(deg, NN);
  k_deg<<<eblk, 256, 0, stream>>>(src, deg, E);
  k_dinv<<<dim3((NN + 255) / 256), 256, 0, stream>>>(deg, NN);
  k_edgew<<<eblk, 256, 0, stream>>>(src, dst, deg, ew, E);
  zerol(h, (size_t)NN * CH);

  // ---- sequential GRU scan over T ----
  for (int t = 0; t < TT; ++t) {
    const float* Xt = X + (size_t)t * NN * CIN;
    zerol(AxAh, (size_t)NN * (CIN + CH));
    k_agg<<<dim3(eblk.x, CIN / 4), 256, 0, stream>>>(src, dst, ew, Xt, Ax, CIN, E);
    k_agg<<<dim3(eblk.x, CH / 4), 256, 0, stream>>>(src, dst, ew, h, Ah, CH, E);
    k_gate_zr<<<ggrid, GTPB, 0, stream>>>(Xt, Ax, h, Ah, Wxz, bxz, Whz, bhz,
                                          Wxr, bxr, Whr, bhr, Zb, HRb);
    zerol(Ahr, (size_t)NN * CH);
    k_agg<<<dim3(eblk.x, CH / 4), 256, 0, stream>>>(src, dst, ew, HRb, Ahr, CH, E);
    k_gate_h<<<ggrid, GTPB, 0, stream>>>(Xt, Ax, h, HRb, Ahr, Wxh, bxh, Whh, bhh, Zb);
    k_fc<<<ggrid, GTPB, 0, stream>>>(h, fcw, fcb, out + (size_t)t * NN * COUT);
  }
}